// IndelCrossAttention_45372034515175
// MI455X (gfx1250) — compile-verified
//
#include <hip/hip_runtime.h>
#include <hip/hip_bf16.h>
#include <math.h>

typedef __attribute__((ext_vector_type(2))) float v2f;
typedef __attribute__((ext_vector_type(8))) float v8f;

#define DIMD   512
#define NHEAD  8
#define HD     64
#define BATCH  4
#define NQ     2048
#define NKEY   2048

// LDS row strides chosen for bank-conflict-free WMMA fragment access
// (64 banks x 4B): stride 68 -> +4 bank rotation per row (b64 frag reads
// disjoint mod-4 classes per lane half); stride 72 -> +8 rotation (b32 V
// frag reads give lane halves disjoint 16-bank windows).
#define KS_STRIDE 68
#define VS_STRIDE 72
#define PS_STRIDE 68
#define NCHUNK   (NKEY / 64)

#if defined(__gfx1250__) && \
    __has_builtin(__builtin_amdgcn_global_load_async_to_lds_b128) && \
    __has_builtin(__builtin_amdgcn_s_wait_asynccnt)
#define USE_ASYNC 1
#else
#define USE_ASYNC 0
#endif

#if USE_ASYNC
// Builtin prototype (from hipcc diagnostics): first arg is int4* in the
// global (__device__, AS1) address space, second is LDS (AS3) int4*.
typedef int async_i4 __attribute__((ext_vector_type(4)));
typedef __attribute__((address_space(1))) async_i4* g_i4p;
typedef __attribute__((address_space(3))) async_i4* l_i4p;
#endif

static __device__ __forceinline__ v8f wmma_f32(v2f a, v2f b, v8f c) {
  // D = A(16x4 f32) * B(4x16 f32) + C(16x16 f32)
  return __builtin_amdgcn_wmma_f32_16x16x4_f32(
      /*neg_a=*/false, a, /*neg_b=*/false, b,
      /*c_mod=*/(short)0, c, /*reuse_a=*/false, /*reuse_b=*/false);
}

// ---------------------------------------------------------------------------
// Generic GEMM: Y[M,N] = X[M,K] * op(W) + bias
//   TRANS_W = true  : op(W)[k,n] = W[n,k]   (torch Linear, W stored (out,in))
//   TRANS_W = false : op(W)[k,n] = W[k,n]
// Block: 128 threads (4 waves). Wave tile: 32 (M) x 64 (N). Grid: (M/128, N/64)
// ---------------------------------------------------------------------------
template <bool TRANS_W>
__global__ __launch_bounds__(128) void gemm_bias_kernel(
    const float* __restrict__ X, const float* __restrict__ Wm,
    const float* __restrict__ bias, float* __restrict__ Y,
    int M, int N, int K) {
  const int lane  = threadIdx.x & 31;
  const int wave  = threadIdx.x >> 5;
  const int mrow  = lane & 15;          // M (for A) / N (for B,C) lane position
  const int klo   = (lane >> 4) * 2;    // K sub-offset for A/B fragments
  const int rbase = (lane >> 4) * 8;    // C/D row offset for upper lane half
  const int m0 = blockIdx.x * 128 + wave * 32;
  const int n0 = blockIdx.y * 64;

  v8f acc[2][4];
#pragma unroll
  for (int nt = 0; nt < 4; ++nt) {
    float bv = bias ? bias[n0 + nt * 16 + mrow] : 0.0f;
    v8f c = {};
#pragma unroll
    for (int r = 0; r < 8; ++r) c[r] = bv;
    acc[0][nt] = c;
    acc[1][nt] = c;
  }

  for (int k0 = 0; k0 < K; k0 += 4) {
    v2f a[2];
#pragma unroll
    for (int mt = 0; mt < 2; ++mt)
      a[mt] = *(const v2f*)(X + (size_t)(m0 + mt * 16 + mrow) * K + k0 + klo);

    v2f b[4];
#pragma unroll
    for (int nt = 0; nt < 4; ++nt) {
      const int col = n0 + nt * 16 + mrow;
      if (TRANS_W) {
        b[nt] = *(const v2f*)(Wm + (size_t)col * K + k0 + klo);
      } else {
        v2f t;
        t[0] = Wm[(size_t)(k0 + klo) * N + col];
        t[1] = Wm[(size_t)(k0 + klo + 1) * N + col];
        b[nt] = t;
      }
    }

#pragma unroll
    for (int mt = 0; mt < 2; ++mt)
#pragma unroll
      for (int nt = 0; nt < 4; ++nt)
        acc[mt][nt] = wmma_f32(a[mt], b[nt], acc[mt][nt]);
  }

#pragma unroll
  for (int mt = 0; mt < 2; ++mt)
#pragma unroll
    for (int nt = 0; nt < 4; ++nt)
#pragma unroll
      for (int r = 0; r < 8; ++r) {
        const int row = m0 + mt * 16 + r + rbase;
        const int col = n0 + nt * 16 + mrow;
        Y[(size_t)row * N + col] = acc[mt][nt][r];
      }
}

// ---------------------------------------------------------------------------
// Stage one 64-key K/V chunk into (padded) LDS buffers. 128 threads, each
// moves 8 float4 per matrix. Async path uses CDNA5 global->LDS async copies
// tracked by ASYNCcnt; fallback is plain b128 load + ds store.
// ---------------------------------------------------------------------------
static __device__ __forceinline__ void stage_chunk(
    const float* __restrict__ Kh, const float* __restrict__ Vh, int kb,
    float* kbuf, float* vbuf, int tid) {
  for (int i = tid; i < 1024; i += 128) {
    const int row = i >> 4;
    const int c4  = (i & 15) * 4;
    const float* gk = Kh + (size_t)(kb + row) * DIMD + c4;
    const float* gv = Vh + (size_t)(kb + row) * DIMD + c4;
#if USE_ASYNC
    __builtin_amdgcn_global_load_async_to_lds_b128(
        (g_i4p)gk, (l_i4p)&kbuf[row * KS_STRIDE + c4], 0, 0);
    __builtin_amdgcn_global_load_async_to_lds_b128(
        (g_i4p)gv, (l_i4p)&vbuf[row * VS_STRIDE + c4], 0, 0);
#else
    *(float4*)&kbuf[row * KS_STRIDE + c4] = *(const float4*)gk;
    *(float4*)&vbuf[row * VS_STRIDE + c4] = *(const float4*)gv;
#endif
  }
}

// ---------------------------------------------------------------------------
// Flash attention over one (batch, head, 64-query tile) per block.
// 4 waves; each wave owns 16 query rows. Keys streamed in 64-wide chunks,
// double-buffered in LDS; online softmax; P converted C->A layout via LDS.
// ---------------------------------------------------------------------------
__global__ __launch_bounds__(128) void attn_kernel(
    const float* __restrict__ Qb, const float* __restrict__ Kb,
    const float* __restrict__ Vb, float* __restrict__ Ob) {
  __shared__ float Ks[2][64 * KS_STRIDE];
  __shared__ float Vs[2][64 * VS_STRIDE];
  __shared__ float Ps[4 * 16 * PS_STRIDE];

  const int lane  = threadIdx.x & 31;
  const int wave  = threadIdx.x >> 5;
  const int mrow  = lane & 15;
  const int klo   = (lane >> 4) * 2;
  const int rbase = (lane >> 4) * 8;

  const int bh = blockIdx.y;
  const int b  = bh >> 3;   // / NHEAD
  const int h  = bh & 7;
  const int q0 = blockIdx.x * 64 + wave * 16;

  const float* Qh = Qb + (size_t)b * NQ * DIMD + h * HD;
  const float* Kh = Kb + (size_t)b * NKEY * DIMD + h * HD;
  const float* Vh = Vb + (size_t)b * NKEY * DIMD + h * HD;
  float* Oh       = Ob + (size_t)b * NQ * DIMD + h * HD;

  // Preload Q fragments for all 16 K-steps over head-dim, fold in 1/sqrt(64)
  v2f qa[16];
#pragma unroll
  for (int ks = 0; ks < 16; ++ks) {
    v2f t = *(const v2f*)(Qh + (size_t)(q0 + mrow) * DIMD + ks * 4 + klo);
    qa[ks] = t * 0.125f;
  }

  v8f o[4];
#pragma unroll
  for (int d = 0; d < 4; ++d) o[d] = (v8f){};
  float run_m[8], run_l[8];
#pragma unroll
  for (int r = 0; r < 8; ++r) { run_m[r] = -3.0e38f; run_l[r] = 0.0f; }

  float* Pw = Ps + wave * 16 * PS_STRIDE;

  stage_chunk(Kh, Vh, 0, Ks[0], Vs[0], threadIdx.x);

  for (int c = 0; c < NCHUNK; ++c) {
    float* kbuf = Ks[c & 1];
    float* vbuf = Vs[c & 1];

#if USE_ASYNC
    if (c + 1 < NCHUNK) {
      // Issue next chunk early; overlap with this chunk's WMMA work.
      stage_chunk(Kh, Vh, (c + 1) * 64, Ks[(c + 1) & 1], Vs[(c + 1) & 1],
                  threadIdx.x);
      __builtin_amdgcn_s_wait_asynccnt(16);  // chunk c (16 ops/wave) complete
    } else {
      __builtin_amdgcn_s_wait_asynccnt(0);
    }
#endif
    __syncthreads();  // all waves' chunk-c data visible in LDS

    // S = (Q/8) * K^T : 16 q-rows x 64 keys
    v8f s[4];
#pragma unroll
    for (int nt = 0; nt < 4; ++nt) s[nt] = (v8f){};
#pragma unroll
    for (int ks = 0; ks < 16; ++ks)
#pragma unroll
      for (int nt = 0; nt < 4; ++nt) {
        v2f bf = *(const v2f*)&kbuf[(nt * 16 + mrow) * KS_STRIDE + ks * 4 + klo];
        s[nt] = wmma_f32(qa[ks], bf, s[nt]);
      }

    // Online softmax: per-row max across 64 keys (rows live in lane halves)
    float nm[8], sc[8];
#pragma unroll
    for (int r = 0; r < 8; ++r) {
      float mx = fmaxf(fmaxf(s[0][r], s[1][r]), fmaxf(s[2][r], s[3][r]));
#pragma unroll
      for (int off = 8; off >= 1; off >>= 1)
        mx = fmaxf(mx, __shfl_xor(mx, off));
      nm[r] = fmaxf(run_m[r], mx);
      sc[r] = __expf(run_m[r] - nm[r]);
    }

    // P = exp(S - nm); write to per-wave LDS slab; accumulate row sums
    float rs[8];
#pragma unroll
    for (int r = 0; r < 8; ++r) rs[r] = 0.0f;
#pragma unroll
    for (int nt = 0; nt < 4; ++nt)
#pragma unroll
      for (int r = 0; r < 8; ++r) {
        float p = __expf(s[nt][r] - nm[r]);
        rs[r] += p;
        Pw[(r + rbase) * PS_STRIDE + nt * 16 + mrow] = p;
      }
#pragma unroll
    for (int r = 0; r < 8; ++r) {
      float t = rs[r];
#pragma unroll
      for (int off = 8; off >= 1; off >>= 1)
        t += __shfl_xor(t, off);
      run_l[r] = run_l[r] * sc[r] + t;
      run_m[r] = nm[r];
    }
#pragma unroll
    for (int d = 0; d < 4; ++d)
#pragma unroll
      for (int r = 0; r < 8; ++r) o[d][r] *= sc[r];

    __syncthreads();  // make P (C-layout) visible before A-layout reads

    // O += P * V   (contract over 64 keys, 16 WMMA K-steps)
#pragma unroll
    for (int ks = 0; ks < 16; ++ks) {
      v2f af = *(const v2f*)&Pw[mrow * PS_STRIDE + ks * 4 + klo];
#pragma unroll
      for (int d = 0; d < 4; ++d) {
        v2f bf;
        bf[0] = vbuf[(ks * 4 + klo) * VS_STRIDE + d * 16 + mrow];
        bf[1] = vbuf[(ks * 4 + klo + 1) * VS_STRIDE + d * 16 + mrow];
        o[d] = wmma_f32(af, bf, o[d]);
      }
    }

#if !USE_ASYNC
    // Sync fallback: stage next chunk after PV so the visibility barrier
    // below (not the compute above) absorbs the global-load latency.
    if (c + 1 < NCHUNK)
      stage_chunk(Kh, Vh, (c + 1) * 64, Ks[(c + 1) & 1], Vs[(c + 1) & 1],
                  threadIdx.x);
#endif
    __syncthreads();  // protect buffers before next chunk's staging/reads
  }

  // Normalize and store
#pragma unroll
  for (int d = 0; d < 4; ++d)
#pragma unroll
    for (int r = 0; r < 8; ++r) {
      float val = o[d][r] / run_l[r];
      Oh[(size_t)(q0 + r + rbase) * DIMD + d * 16 + mrow] = val;
    }
}

// ---------------------------------------------------------------------------
// score[row] = softplus(scale * dot(weighted[row,:], context[row,:]))
// One wave per row.
// ---------------------------------------------------------------------------
__global__ __launch_bounds__(128) void score_kernel(
    const float* __restrict__ Ctx, const float* __restrict__ Wt,
    const float* __restrict__ scale, float* __restrict__ out, int rows) {
  const int lane = threadIdx.x & 31;
  const int wave = threadIdx.x >> 5;
  const int row  = blockIdx.x * 4 + wave;
  if (row >= rows) return;
  const float* cp = Ctx + (size_t)row * DIMD;
  const float* wp = Wt + (size_t)row * DIMD;
  float acc = 0.0f;
#pragma unroll
  for (int i = lane * 4; i < DIMD; i += 128) {
    float4 a = *(const float4*)(cp + i);
    float4 b = *(const float4*)(wp + i);
    acc += a.x * b.x + a.y * b.y + a.z * b.z + a.w * b.w;
  }
#pragma unroll
  for (int off = 16; off >= 1; off >>= 1)
    acc += __shfl_xor(acc, off);
  if (lane == 0) {
    float s = acc * scale[0];
    out[row] = (s > 20.0f) ? s : log1pf(__expf(s));
  }
}

// ---------------------------------------------------------------------------
extern "C" void kernel_launch(void* const* d_in, const int* in_sizes, int n_in,
                              void* d_out, int out_size, void* d_ws, size_t ws_size,
                              hipStream_t stream) {
  (void)in_sizes; (void)n_in; (void)out_size; (void)ws_size;
  const float* g1 = (const float*)d_in[0];
  const float* g2 = (const float*)d_in[1];
  const float* Wq = (const float*)d_in[2];
  const float* bq = (const float*)d_in[3];
  const float* Wk = (const float*)d_in[4];
  const float* bk = (const float*)d_in[5];
  const float* Wv = (const float*)d_in[6];
  const float* bv = (const float*)d_in[7];
  const float* Wo = (const float*)d_in[8];
  const float* bo = (const float*)d_in[9];
  const float* W  = (const float*)d_in[10];
  const float* scale = (const float*)d_in[11];
  float* out = (float*)d_out;

  const size_t SLAB = (size_t)BATCH * NQ * DIMD;  // floats per intermediate
  float* ws = (float*)d_ws;
  float* Qw = ws;
  float* Kw = ws + SLAB;
  float* Vw = ws + 2 * SLAB;
  float* Ow = ws + 3 * SLAB;
  float* Wt = ws + 4 * SLAB;
  float* Cx = Qw;  // context reuses Q slab (Q dead after attention)

  const int M = BATCH * NQ;  // 8192 rows (N1 == N2)
  dim3 gg(M / 128, DIMD / 64);
  gemm_bias_kernel<true><<<gg, 128, 0, stream>>>(g1, Wq, bq, Qw, M, DIMD, DIMD);
  gemm_bias_kernel<true><<<gg, 128, 0, stream>>>(g2, Wk, bk, Kw, M, DIMD, DIMD);
  gemm_bias_kernel<true><<<gg, 128, 0, stream>>>(g2, Wv, bv, Vw, M, DIMD, DIMD);
  gemm_bias_kernel<false><<<gg, 128, 0, stream>>>(g1, W, nullptr, Wt, M, DIMD, DIMD);

  dim3 ga(NQ / 64, BATCH * NHEAD);
  attn_kernel<<<ga, 128, 0, stream>>>(Qw, Kw, Vw, Ow);

  gemm_bias_kernel<true><<<gg, 128, 0, stream>>>(Ow, Wo, bo, Cx, M, DIMD, DIMD);

  score_kernel<<<(M + 3) / 4, 128, 0, stream>>>(Cx, Wt, scale, out, M);
}